// Voxelization_57397942944138
// MI455X (gfx1250) — compile-verified
//
#include <hip/hip_runtime.h>
#include <hip/hip_bf16.h>
#include <cstdint>

// ---------------- problem constants (from the reference) ----------------
#define GX 800
#define GY 800
#define GZ 160
#define NCELLS (GX * GY * GZ)               // 102,400,000
#define NWORDS (NCELLS / 32)                // 3,200,000 (exact)
#define SCAN_BLOCKS (NWORDS / 256)          // 12,500 (exact)
#define MAXV 60000
#define MAXP 32
#define PMINX (-20.0f)
#define PMINY (-20.0f)
#define PMINZ (-2.0f)
#define VS 0.05f

// d_out byte offsets (voxels f32[60000*4*32], indices i64[60000*3], num_points i64[60000])
#define VOX_FLOATS (MAXV * 4 * MAXP)        // 7,680,000
#define IDX_BYTE_OFF ((size_t)VOX_FLOATS * 4)        // 30,720,000
#define NP_BYTE_OFF (IDX_BYTE_OFF + (size_t)MAXV * 3 * 8) // 32,160,000
#define OUT_BYTES (NP_BYTE_OFF + (size_t)MAXV * 8)   // 32,640,000

// ---------------- tiny memset ----------------
__global__ void zero_u4_kernel(uint4* __restrict__ p, long long n4) {
    long long i = (long long)blockIdx.x * blockDim.x + threadIdx.x;
    long long stride = (long long)gridDim.x * blockDim.x;
    const uint4 z = make_uint4(0u, 0u, 0u, 0u);
    for (; i < n4; i += stride) p[i] = z;
}

// ---------------- pass 1: mark occupancy bitmap ----------------
// Streams the point cloud through LDS with the CDNA5 async data mover
// (global_load_async_to_lds_b128, ASYNCcnt), then each lane consumes its own
// staged float4. Same-wave producer/consumer, so s_wait_asynccnt 0 suffices
// (no barrier needed).
__global__ void mark_kernel(const float4* __restrict__ pts,
                            unsigned int* __restrict__ bitmap, int n) {
    __shared__ float4 tile[256];
    int idx = blockIdx.x * 256 + threadIdx.x;
    if (idx < n) {
        unsigned long long gaddr = (unsigned long long)(const void*)(pts + idx);
        // low 32 bits of the generic shared address == LDS byte offset
        unsigned int laddr = (unsigned int)(size_t)(&tile[threadIdx.x]);
        asm volatile("global_load_async_to_lds_b128 %0, %1, off"
                     :: "v"(laddr), "v"(gaddr) : "memory");
    }
    asm volatile("s_wait_asynccnt 0" ::: "memory");
    if (idx >= n) return;
    float4 p = tile[threadIdx.x];

    float fx = (p.x - PMINX) / VS;
    float fy = (p.y - PMINY) / VS;
    float fz = (p.z - PMINZ) / VS;
    int cx = (int)fx, cy = (int)fy, cz = (int)fz;   // trunc-toward-zero == astype(int32)
    if (cx < 0 || cx >= GX || cy < 0 || cy >= GY || cz < 0 || cz >= GZ) return;
    unsigned int lin = (unsigned int)cx + (unsigned int)cy * GX
                     + (unsigned int)cz * (GX * GY);
    atomicOr(&bitmap[lin >> 5], 1u << (lin & 31u));
}

// ---------------- pass 2a: per-word popcount, block-exclusive scan ----------------
__global__ void word_scan_kernel(const unsigned int* __restrict__ bitmap,
                                 unsigned int* __restrict__ wordPrefix,
                                 unsigned int* __restrict__ blockSums, int nwords) {
    __shared__ unsigned int s[256];
    int i = blockIdx.x * 256 + threadIdx.x;
    unsigned int pc = (i < nwords) ? (unsigned int)__popc(bitmap[i]) : 0u;
    s[threadIdx.x] = pc;
    __syncthreads();
    for (int off = 1; off < 256; off <<= 1) {
        unsigned int t = (threadIdx.x >= (unsigned)off) ? s[threadIdx.x - off] : 0u;
        __syncthreads();
        s[threadIdx.x] += t;
        __syncthreads();
    }
    if (i < nwords) wordPrefix[i] = s[threadIdx.x] - pc;   // exclusive within block
    if (threadIdx.x == 255) blockSums[blockIdx.x] = s[255];
}

// ---------------- pass 2b: exclusive scan of block sums (single block) ----------------
__global__ void scan_block_sums_kernel(unsigned int* __restrict__ bs, int n) {
    __shared__ unsigned int s[1024];
    unsigned int carry = 0u;
    for (int base = 0; base < n; base += 1024) {
        int i = base + (int)threadIdx.x;
        unsigned int v = (i < n) ? bs[i] : 0u;
        s[threadIdx.x] = v;
        __syncthreads();
        for (int off = 1; off < 1024; off <<= 1) {
            unsigned int t = (threadIdx.x >= (unsigned)off) ? s[threadIdx.x - off] : 0u;
            __syncthreads();
            s[threadIdx.x] += t;
            __syncthreads();
        }
        unsigned int incl = s[threadIdx.x];
        if (i < n) bs[i] = (incl - v) + carry;             // exclusive + carry
        unsigned int total = s[1023];
        __syncthreads();
        carry += total;
    }
}

// ---------------- pass 2c: add scanned block offsets ----------------
__global__ void add_offsets_kernel(unsigned int* __restrict__ wordPrefix,
                                   const unsigned int* __restrict__ bs, int nwords) {
    int i = blockIdx.x * 256 + threadIdx.x;
    if (i < nwords) wordPrefix[i] += bs[blockIdx.x];
}

// ---------------- pass 3: scatter points into voxels ----------------
__global__ void scatter_kernel(const float4* __restrict__ pts,
                               const unsigned int* __restrict__ bitmap,
                               const unsigned int* __restrict__ wordPrefix,
                               unsigned int* __restrict__ counts,
                               float* __restrict__ voxels,
                               long long* __restrict__ indices, int n) {
    int idx = blockIdx.x * blockDim.x + threadIdx.x;
    if (idx >= n) return;
    float4 p = pts[idx];
    float fx = (p.x - PMINX) / VS;
    float fy = (p.y - PMINY) / VS;
    float fz = (p.z - PMINZ) / VS;
    int cx = (int)fx, cy = (int)fy, cz = (int)fz;
    if (cx < 0 || cx >= GX || cy < 0 || cy >= GY || cz < 0 || cz >= GZ) return;
    unsigned int lin = (unsigned int)cx + (unsigned int)cy * GX
                     + (unsigned int)cz * (GX * GY);
    unsigned int w = lin >> 5, b = lin & 31u;
    unsigned int word = bitmap[w];                          // L2-resident (12.8 MB)
    unsigned int rank = wordPrefix[w]
                      + (unsigned int)__popc(word & ((1u << b) - 1u));
    if (rank >= (unsigned int)MAXV) return;                 // keep 60000 lowest linear ids
    unsigned int slot = atomicAdd(&counts[rank], 1u);
    if (slot == 0u) {                                       // one writer per voxel
        indices[(size_t)rank * 3 + 0] = (long long)cx;
        indices[(size_t)rank * 3 + 1] = (long long)cy;
        indices[(size_t)rank * 3 + 2] = (long long)cz;
    }
    if (slot < (unsigned int)MAXP) {
        // layout [V, C=4, 32]: element (v,c,s) at v*128 + c*32 + s
        float* vb = voxels + (size_t)rank * (4 * MAXP) + slot;
        vb[0 * MAXP] = p.x;
        vb[1 * MAXP] = p.y;
        vb[2 * MAXP] = p.z;
        vb[3 * MAXP] = p.w;
    }
}

// ---------------- pass 4: num_points = min(count, 32) as int64 ----------------
__global__ void finalize_kernel(const unsigned int* __restrict__ counts,
                                long long* __restrict__ np) {
    int v = blockIdx.x * 256 + threadIdx.x;
    if (v < MAXV) {
        unsigned int c = counts[v];
        np[v] = (long long)(c > (unsigned)MAXP ? (unsigned)MAXP : c);
    }
}

extern "C" void kernel_launch(void* const* d_in, const int* in_sizes, int n_in,
                              void* d_out, int out_size, void* d_ws, size_t ws_size,
                              hipStream_t stream) {
    (void)n_in; (void)out_size; (void)ws_size;
    const float4* pts = (const float4*)d_in[0];
    const int n = in_sizes[0] / 4;                          // 4 channels per point

    // workspace layout (u32 units)
    unsigned int* bitmap     = (unsigned int*)d_ws;         // NWORDS
    unsigned int* wordPrefix = bitmap + NWORDS;             // NWORDS
    unsigned int* blockSums  = wordPrefix + NWORDS;         // SCAN_BLOCKS (padded)
    unsigned int* counts     = blockSums + 12544;           // MAXV

    float*     out_voxels  = (float*)d_out;
    long long* out_indices = (long long*)((char*)d_out + IDX_BYTE_OFF);
    long long* out_np      = (long long*)((char*)d_out + NP_BYTE_OFF);

    // 0) zero outputs + bitmap + counters (all sizes are multiples of 16 B)
    zero_u4_kernel<<<2048, 256, 0, stream>>>((uint4*)d_out,
                                             (long long)(OUT_BYTES / 16));
    zero_u4_kernel<<<2048, 256, 0, stream>>>((uint4*)bitmap,
                                             (long long)(NWORDS * 4 / 16));
    zero_u4_kernel<<<64, 256, 0, stream>>>((uint4*)counts,
                                           (long long)(MAXV * 4 / 16));

    int ptBlocks = (n + 255) / 256;

    // 1) occupancy bitmap (async global->LDS staging)
    mark_kernel<<<ptBlocks, 256, 0, stream>>>(pts, bitmap, n);

    // 2) exclusive prefix over per-word popcounts -> dense voxel ranks
    word_scan_kernel<<<SCAN_BLOCKS, 256, 0, stream>>>(bitmap, wordPrefix,
                                                      blockSums, NWORDS);
    scan_block_sums_kernel<<<1, 1024, 0, stream>>>(blockSums, SCAN_BLOCKS);
    add_offsets_kernel<<<SCAN_BLOCKS, 256, 0, stream>>>(wordPrefix, blockSums,
                                                        NWORDS);

    // 3) scatter points into [V,4,32] voxel buffer + write indices
    scatter_kernel<<<ptBlocks, 256, 0, stream>>>(pts, bitmap, wordPrefix, counts,
                                                 out_voxels, out_indices, n);

    // 4) num_points
    finalize_kernel<<<(MAXV + 255) / 256, 256, 0, stream>>>(counts, out_np);
}